// Model_89343909692066
// MI455X (gfx1250) — compile-verified
//
#include <hip/hip_runtime.h>
#include <math.h>

// ---------------- problem constants (match reference) ----------------
#define BGR      128          // batch of graphs
#define N_PER    1024         // nodes per graph
#define NNODES   (BGR * N_PER)
#define DEGC     16           // edges per node
#define EPG      (N_PER * DEGC)   // 16384 edges per graph, contiguous block
#define F_IN     64
#define TOTAL_D  97           // 32+32+32+1 concat width
#define TOPK     30
#define NC1      16
#define NC2      32
#define KW2      5
#define OUT_T    11           // (30/2) - 5 + 1
#define HALFN    512          // nodes handled per block (graph split in 2)
#define PSTRIDE  65           // LDS row stride (65 mod 64 == 1 -> conflict free)
#define NTHREADS 256
#define NW       8            // waves per block (wave32)

typedef __attribute__((ext_vector_type(2))) float v2f;
typedef __attribute__((ext_vector_type(8))) float v8f;

// ---------------- degree: per-graph LDS histogram ----------------
__global__ __launch_bounds__(NTHREADS) void deg_kernel(
    const int* __restrict__ edst, float* __restrict__ deg)
{
    __shared__ int s_cnt[N_PER];
    const int g = blockIdx.x, tid = threadIdx.x;
    const int gbase = g * N_PER, ebase = g * EPG;
    for (int i = tid; i < N_PER; i += NTHREADS) s_cnt[i] = 0;
    __syncthreads();
    for (int e = ebase + tid; e < ebase + EPG; e += NTHREADS)
        atomicAdd(&s_cnt[edst[e] - gbase], 1);
    __syncthreads();
    for (int i = tid; i < N_PER; i += NTHREADS)
        deg[gbase + i] = (float)s_cnt[i] + 1.0f;
}

// ---------------- fused GCN layer: scatter (LDS atomics) + WMMA GEMM ----------------
// Templated on D_IN/D_OUT so all inner loops fully unroll and the per-wave
// B fragments live in VGPRs (reused across all 8 M-tiles of the wave).
template <int D_IN, int D_OUT>
__global__ __launch_bounds__(NTHREADS) void gcn_layer_kernel(
    const float* __restrict__ h_in, int in_stride,
    const float* __restrict__ W, const float* __restrict__ bias,
    float* __restrict__ h_out,          // points into cat at this layer's column offset
    const float* __restrict__ deg,
    const int* __restrict__ esrc, const int* __restrict__ edst)
{
    __shared__ float s_pool[HALFN * PSTRIDE];   // 133,120 B
    __shared__ float s_w[D_IN * D_OUT];
    __shared__ float s_b[D_OUT];
    __shared__ float s_deg[HALFN];

    const int tid  = threadIdx.x;
    const int g    = blockIdx.x >> 1;
    const int half = blockIdx.x & 1;
    const int gfirst = g * N_PER + half * HALFN;  // first global node of this block
    const int ebase  = g * EPG;
    const int lane = tid & 31, wv = tid >> 5;

    for (int i = tid; i < HALFN * PSTRIDE; i += NTHREADS) s_pool[i] = 0.0f;
    for (int i = tid; i < D_IN * D_OUT; i += NTHREADS)    s_w[i] = W[i];
    if (tid < D_OUT) s_b[tid] = bias[tid];
    for (int i = tid; i < HALFN; i += NTHREADS)           s_deg[i] = deg[gfirst + i];
    __syncthreads();

    // ---- edge scatter: coalesced edge reads, shfl-broadcast, ds_add_f32 ----
    constexpr int NCHUNK = EPG / (NW * 32);               // 64 chunks per block
    for (int c = 0; c < NCHUNK; ++c) {
        const int e = ebase + (c * NW + wv) * 32 + lane;
        const int mdst = edst[e] - gfirst;
        const int msrc = esrc[e];
        // warm L2 for all 32 source-feature rows of this chunk (global_prefetch_b8)
        __builtin_prefetch(h_in + (size_t)msrc * in_stride, 0, 2);
        if (c + 1 < NCHUNK) {                             // prefetch next edge chunk
            __builtin_prefetch(&edst[e + NW * 32], 0, 2);
            __builtin_prefetch(&esrc[e + NW * 32], 0, 2);
        }
        for (int j = 0; j < 32; ++j) {
            const int dl = __shfl(mdst, j, 32);           // uniform per wave
            if ((unsigned)dl < (unsigned)HALFN) {
                const int src = __shfl(msrc, j, 32);
                const float* hs = h_in + (size_t)src * in_stride;
#pragma unroll
                for (int k = lane; k < D_IN; k += 32)
                    atomicAdd(&s_pool[dl * PSTRIDE + k], hs[k]);  // ds_add_f32
            }
        }
    }
    __syncthreads();

    // ---- self term: pooled += h ----
    for (int i = tid; i < HALFN * D_IN; i += NTHREADS) {
        const int n = i / D_IN;                            // D_IN is power of two
        const int k = i % D_IN;
        s_pool[n * PSTRIDE + k] += h_in[(size_t)(gfirst + n) * in_stride + k];
    }
    __syncthreads();

    if constexpr (D_OUT == 32) {
        // 32 M-tiles x 2 N-tiles over 8 waves: each wave owns a fixed N-tile
        // so its B fragments are loaded once into VGPRs and reused 8x.
        const int lm = lane & 15, hi = lane >> 4;
        const int n0 = (wv & 1) * 16;
        constexpr int KQ = D_IN / 4;                       // 16 or 8 wmma steps
        v2f bf[KQ];
#pragma unroll
        for (int k = 0; k < KQ; ++k) {
            const int ka = 4 * k + 2 * hi;                 // lanes 16-31 hold K=2,3
            bf[k].x = s_w[ka * D_OUT + n0 + lm];
            bf[k].y = s_w[(ka + 1) * D_OUT + n0 + lm];
        }
        const float bv = s_b[n0 + lm];
#pragma unroll 2
        for (int mt = (wv >> 1); mt < HALFN / 16; mt += NW / 2) {
            const int m0 = mt * 16;
            v8f acc = {0.f, 0.f, 0.f, 0.f, 0.f, 0.f, 0.f, 0.f};
#pragma unroll
            for (int k = 0; k < KQ; ++k) {
                const int ka = 4 * k + 2 * hi;
                v2f a;
                a.x = s_pool[(m0 + lm) * PSTRIDE + ka];
                a.y = s_pool[(m0 + lm) * PSTRIDE + ka + 1];
                acc = __builtin_amdgcn_wmma_f32_16x16x4_f32(
                    false, a, false, bf[k], (short)0, acc, false, false);
            }
#pragma unroll
            for (int r = 0; r < 8; ++r) {
                const int nl = m0 + r + 8 * hi;            // lanes 16-31 hold M=8..15
                const float val = tanhf((acc[r] + bv) / s_deg[nl]);
                h_out[(size_t)(gfirst + nl) * TOTAL_D + (n0 + lm)] = val;
            }
        }
    } else {
        // final layer: D_OUT == 1 -> plain GEMV
        for (int n = tid; n < HALFN; n += NTHREADS) {
            float s = 0.0f;
#pragma unroll
            for (int k = 0; k < D_IN; ++k) s += s_pool[n * PSTRIDE + k] * s_w[k];
            h_out[(size_t)(gfirst + n) * TOTAL_D] = tanhf((s + s_b[0]) / s_deg[n]);
        }
    }
}

// ---------------- top-30 selection per graph (descending, ties -> lower index) ----------------
__global__ __launch_bounds__(NTHREADS) void topk_kernel(
    const float* __restrict__ cat, int* __restrict__ topk)
{
    __shared__ float s_val[N_PER];
    __shared__ float rv[NTHREADS];
    __shared__ int   ri[NTHREADS];
    const int g = blockIdx.x, tid = threadIdx.x;
    for (int i = tid; i < N_PER; i += NTHREADS)
        s_val[i] = cat[(size_t)(g * N_PER + i) * TOTAL_D + (TOTAL_D - 1)];
    __syncthreads();
    for (int it = 0; it < TOPK; ++it) {
        float bv = -1e30f; int bi = N_PER;
        for (int i = tid; i < N_PER; i += NTHREADS) {
            const float v = s_val[i];
            if (v > bv || (v == bv && i < bi)) { bv = v; bi = i; }
        }
        rv[tid] = bv; ri[tid] = bi;
        __syncthreads();
        for (int s = NTHREADS / 2; s > 0; s >>= 1) {
            if (tid < s) {
                const float ov = rv[tid + s]; const int oi = ri[tid + s];
                if (ov > rv[tid] || (ov == rv[tid] && oi < ri[tid])) { rv[tid] = ov; ri[tid] = oi; }
            }
            __syncthreads();
        }
        if (tid == 0) { topk[g * TOPK + it] = ri[0]; s_val[ri[0]] = -1e30f; }
        __syncthreads();
    }
}

// ---------------- head: gather + conv1(1x1) + ReLU + pairmax + conv2(k=5) + ReLU ----------------
__global__ __launch_bounds__(NTHREADS) void head_kernel(
    const float* __restrict__ cat, const int* __restrict__ topk,
    const float* __restrict__ c1w, const float* __restrict__ c1b,
    const float* __restrict__ c2w, const float* __restrict__ c2b,
    float* __restrict__ out)
{
    __shared__ float s_pg[TOPK * TOTAL_D];   // 30 x 97
    __shared__ float s_x1[NC1 * TOPK];       // 16 x 30
    __shared__ float s_x2[NC1 * (TOPK / 2)]; // 16 x 15
    const int g = blockIdx.x, tid = threadIdx.x;

    for (int i = tid; i < TOPK * TOTAL_D; i += NTHREADS) {
        const int k = i / TOTAL_D, d = i % TOTAL_D;
        const int node = topk[g * TOPK + k];
        s_pg[i] = cat[((size_t)g * N_PER + node) * TOTAL_D + d];
    }
    __syncthreads();
    for (int i = tid; i < NC1 * TOPK; i += NTHREADS) {
        const int c = i / TOPK, k = i % TOPK;
        float s = c1b[c];
        for (int d = 0; d < TOTAL_D; ++d) s += c1w[c * TOTAL_D + d] * s_pg[k * TOTAL_D + d];
        s_x1[c * TOPK + k] = fmaxf(s, 0.0f);
    }
    __syncthreads();
    for (int i = tid; i < NC1 * (TOPK / 2); i += NTHREADS) {
        const int c = i / (TOPK / 2), j = i % (TOPK / 2);
        s_x2[c * (TOPK / 2) + j] = fmaxf(s_x1[c * TOPK + 2 * j], s_x1[c * TOPK + 2 * j + 1]);
    }
    __syncthreads();
    for (int i = tid; i < NC2 * OUT_T; i += NTHREADS) {
        const int co = i / OUT_T, t = i % OUT_T;
        float s = c2b[co];
        for (int ci = 0; ci < NC1; ++ci)
            for (int w = 0; w < KW2; ++w)
                s += c2w[(co * NC1 + ci) * KW2 + w] * s_x2[ci * (TOPK / 2) + t + w];
        out[(size_t)g * (NC2 * OUT_T) + i] = fmaxf(s, 0.0f);
    }
}

// ---------------- launcher ----------------
extern "C" void kernel_launch(void* const* d_in, const int* in_sizes, int n_in,
                              void* d_out, int out_size, void* d_ws, size_t ws_size,
                              hipStream_t stream) {
    const float* node_feat = (const float*)d_in[0];
    const int*   esrc      = (const int*)  d_in[1];
    const int*   edst      = (const int*)  d_in[2];
    const float* W0 = (const float*)d_in[3];  const float* b0 = (const float*)d_in[4];
    const float* W1 = (const float*)d_in[5];  const float* b1 = (const float*)d_in[6];
    const float* W2 = (const float*)d_in[7];  const float* b2 = (const float*)d_in[8];
    const float* W3 = (const float*)d_in[9];  const float* b3 = (const float*)d_in[10];
    const float* c1w = (const float*)d_in[11]; const float* c1b = (const float*)d_in[12];
    const float* c2w = (const float*)d_in[13]; const float* c2b = (const float*)d_in[14];

    // workspace layout: deg (N f32) | cat (N x 97 f32) | topk (B*K i32)  ~= 51.4 MB
    float* deg = (float*)d_ws;
    float* cat = deg + NNODES;
    int*   tki = (int*)(cat + (size_t)NNODES * TOTAL_D);

    deg_kernel<<<BGR, NTHREADS, 0, stream>>>(edst, deg);

    // layer 0: node_feat (stride 64, d_in 64) -> cat cols [0,32)
    gcn_layer_kernel<64, 32><<<BGR * 2, NTHREADS, 0, stream>>>(
        node_feat, F_IN, W0, b0, cat + 0, deg, esrc, edst);
    // layer 1: cat[0:32) -> cat cols [32,64)
    gcn_layer_kernel<32, 32><<<BGR * 2, NTHREADS, 0, stream>>>(
        cat + 0, TOTAL_D, W1, b1, cat + 32, deg, esrc, edst);
    // layer 2: cat[32:64) -> cat cols [64,96)
    gcn_layer_kernel<32, 32><<<BGR * 2, NTHREADS, 0, stream>>>(
        cat + 32, TOTAL_D, W2, b2, cat + 64, deg, esrc, edst);
    // layer 3: cat[64:96) -> cat col 96 (D_OUT = 1)
    gcn_layer_kernel<32, 1><<<BGR * 2, NTHREADS, 0, stream>>>(
        cat + 64, TOTAL_D, W3, b3, cat + 96, deg, esrc, edst);

    topk_kernel<<<BGR, NTHREADS, 0, stream>>>(cat, tki);
    head_kernel<<<BGR, NTHREADS, 0, stream>>>(cat, tki, c1w, c1b, c2w, c2b, (float*)d_out);
}